// MyRNNCell_80221399154871
// MI455X (gfx1250) — compile-verified
//
#include <hip/hip_runtime.h>

// ---------------------------------------------------------------------------
// Hopf normal-form ODE, Dormand-Prince 5 fixed step, 96 steps over [0, 2*pi].
// Per stage: y^H C y (complex scalar) and A@y, B@y with dense complex 2048^2
// matrices -> 6 real matmuls M @ [y_re, y_im] via V_WMMA_F32_16X16X4_F32.
// Matrices (100 MB) are L2-resident on MI455X (192 MB L2); the GEMV is an
// L2-latency-bound stream, so K is split 4x for memory-level parallelism
// (3072 waves, per-split partial outputs, combined in the fused update).
// ---------------------------------------------------------------------------

typedef __attribute__((ext_vector_type(2))) float v2f;
typedef __attribute__((ext_vector_type(8))) float v8f;

struct MatPtrs { const float* m[6]; };   // A_re, A_im, B_re, B_im, C_re, C_im

#define WAVES_PER_BLOCK 4
#define MV_BLOCK (WAVES_PER_BLOCK * 32)
#define CHUNK 16                          // WMMAs per chunk (K advance = 64)
#define KSPLIT 4                          // K-dimension split factor

// Gp layout: Gp[split][(mat*2 + col)][n], col 0 = M@y_re slice, col 1 = M@y_im
// LDS: [0,L) y_re slice, [L,2L) y_im slice, [2L,3L+8) zeros (dead-column pool)
__global__ __launch_bounds__(MV_BLOCK)
void hopf_matvec(MatPtrs mats,
                 const float* __restrict__ y_re, const float* __restrict__ y_im,
                 float* __restrict__ Gp, int n, int tiles)
{
    extern __shared__ float lds[];
    const int tid    = threadIdx.x;
    const int split  = blockIdx.y;
    const int L      = n / KSPLIT;
    const int kbegin = split * L;

    for (int t = tid; t < L; t += MV_BLOCK) {
        lds[t]     = y_re[kbegin + t];
        lds[L + t] = y_im[kbegin + t];
    }
    for (int t = tid; t < L + 8; t += MV_BLOCK) {
        lds[2 * L + t] = 0.0f;
    }
    __syncthreads();

    const int wave = tid >> 5;
    const int lane = tid & 31;
    const int tile = blockIdx.x * WAVES_PER_BLOCK + wave;
    if (tile >= tiles) return;                     // whole wave exits together

    const int tiles_per_mat = n >> 4;
    const int mat  = tile / tiles_per_mat;
    const int row0 = (tile % tiles_per_mat) << 4;

    // A operand (16x4 f32): lane<16 holds (row=lane, K, K+1); lane>=16 holds (K+2, K+3)
    const int row  = lane & 15;
    const int koff = (lane >> 4) << 1;             // 0 or 2
    const float* arow = mats.m[mat] + (size_t)(row0 + row) * (size_t)n + kbegin + koff;

    // B operand (4x16 f32): lane n-index = lane&15; only cols 0 (y_re), 1 (y_im) live.
    // All lanes stride identically with k; dead columns walk the zero pool.
    const int col = lane & 15;
    const float* bbase = (col < 2) ? &lds[col * L + koff] : &lds[2 * L + koff];

    v8f acc = {};
    for (int k = 0; k < L; k += 4 * CHUNK) {
        __builtin_prefetch(arow + k + 4 * CHUNK * 4, 0, 3);   // ahead, WGP scope
        v2f a[CHUNK];
        v2f b[CHUNK];
#pragma unroll
        for (int i = 0; i < CHUNK; ++i) {
            a[i] = *(const v2f*)(arow + k + 4 * i);           // global_load_b64
            b[i] = *(const v2f*)(bbase + k + 4 * i);          // ds_load_b64 (paired)
        }
#pragma unroll
        for (int i = 0; i < CHUNK; ++i) {
            acc = __builtin_amdgcn_wmma_f32_16x16x4_f32(
                false, a[i], false, b[i], (short)0, acc, false, false);
        }
    }

    // D (16x16 f32): lane<16 -> v[i] = D[i][lane]; lane>=16 -> v[i] = D[8+i][lane-16]
    if (col < 2) {
        float* dst = Gp + ((size_t)(split * 12 + mat * 2 + col)) * (size_t)n
                        + row0 + ((lane >> 4) << 3);
#pragma unroll
        for (int i = 0; i < 8; ++i) dst[i] = acc[i];
    }
}

// Fused: s = y^H (C y) reduction, then k_stage = A@y - s*(B@y) and
// y_next = z + h * sum_j c_j k_j. Single block; split partials summed inline.
#define DU_BLOCK 1024
__global__ __launch_bounds__(DU_BLOCK)
void hopf_dot_update(const float* __restrict__ Gp,
                     const float* __restrict__ z_re, const float* __restrict__ z_im,
                     float* __restrict__ Kbuf,            // [6 stages][2 (re,im)][n]
                     float* __restrict__ y_re, float* __restrict__ y_im,
                     float* __restrict__ zn_re, float* __restrict__ zn_im,
                     int stage, int ncoef,
                     float c0, float c1, float c2, float c3, float c4, float c5,
                     float h, int write_z, int n)
{
    __shared__ float sre[DU_BLOCK];
    __shared__ float sim[DU_BLOCK];
    const int tid = threadIdx.x;
    const size_t N = (size_t)n;

    // Phase 1: s = sum_k conj(y_k) * (C y)_k, C parts are Gp mats 4 (C_re), 5 (C_im)
    float ar = 0.0f, ai = 0.0f;
    for (int k = tid; k < n; k += DU_BLOCK) {
        float cyr = 0.0f, cyi = 0.0f;
#pragma unroll
        for (int sp = 0; sp < KSPLIT; ++sp) {
            const float* g = Gp + (size_t)(sp * 12) * N;
            cyr += g[8 * N + k] - g[11 * N + k];   // C_re@yr - C_im@yi
            cyi += g[9 * N + k] + g[10 * N + k];   // C_re@yi + C_im@yr
        }
        float yr = y_re[k], yi = y_im[k];
        ar += yr * cyr + yi * cyi;                 // Re(conj(y) * Cy)
        ai += yr * cyi - yi * cyr;                 // Im(conj(y) * Cy)
    }
    sre[tid] = ar;
    sim[tid] = ai;
    __syncthreads();
    for (int off = DU_BLOCK / 2; off > 0; off >>= 1) {
        if (tid < off) {
            sre[tid] += sre[tid + off];
            sim[tid] += sim[tid + off];
        }
        __syncthreads();
    }
    const float sr = sre[0];
    const float si = sim[0];
    __syncthreads();

    // Phase 2: elementwise stage update
    float c[6] = {c0, c1, c2, c3, c4, c5};
    for (int k = tid; k < n; k += DU_BLOCK) {
        float Ayr = 0.0f, Ayi = 0.0f, Byr = 0.0f, Byi = 0.0f;
#pragma unroll
        for (int sp = 0; sp < KSPLIT; ++sp) {
            const float* g = Gp + (size_t)(sp * 12) * N;
            Ayr += g[0 * N + k] - g[3 * N + k];    // A_re@yr - A_im@yi
            Ayi += g[1 * N + k] + g[2 * N + k];    // A_re@yi + A_im@yr
            Byr += g[4 * N + k] - g[7 * N + k];
            Byi += g[5 * N + k] + g[6 * N + k];
        }
        float kr = Ayr - (sr * Byr - si * Byi);
        float ki = Ayi - (sr * Byi + si * Byr);
        Kbuf[(size_t)(stage * 2 + 0) * N + k] = kr;
        Kbuf[(size_t)(stage * 2 + 1) * N + k] = ki;
        float accr = 0.0f, acci = 0.0f;
        for (int j = 0; j < ncoef; ++j) {
            float kjr = (j == stage) ? kr : Kbuf[(size_t)(j * 2 + 0) * N + k];
            float kji = (j == stage) ? ki : Kbuf[(size_t)(j * 2 + 1) * N + k];
            accr += c[j] * kjr;
            acci += c[j] * kji;
        }
        float yr = z_re[k] + h * accr;
        float yi = z_im[k] + h * acci;
        y_re[k] = yr;
        y_im[k] = yi;
        if (write_z) { zn_re[k] = yr; zn_im[k] = yi; }
    }
}

__global__ __launch_bounds__(256)
void hopf_init(const float* __restrict__ zi_re, const float* __restrict__ zi_im,
               float* __restrict__ z_re, float* __restrict__ z_im,
               float* __restrict__ y_re, float* __restrict__ y_im, int n)
{
    int k = blockIdx.x * 256 + threadIdx.x;
    if (k >= n) return;
    float a = zi_re[k], b = zi_im[k];
    z_re[k] = a; z_im[k] = b;
    y_re[k] = a; y_im[k] = b;
}

__global__ __launch_bounds__(256)
void hopf_out(const float* __restrict__ z_re, const float* __restrict__ z_im,
              float* __restrict__ out, int n)
{
    int k = blockIdx.x * 256 + threadIdx.x;
    if (k >= n) return;
    out[k]     = z_re[k];
    out[n + k] = z_im[k];
}

extern "C" void kernel_launch(void* const* d_in, const int* in_sizes, int n_in,
                              void* d_out, int out_size, void* d_ws, size_t ws_size,
                              hipStream_t stream)
{
    const int n = in_sizes[0];                     // 2048
    const float* zin_re = (const float*)d_in[0];
    const float* zin_im = (const float*)d_in[1];
    // d_in[2], d_in[3] (x_re, x_im) are dead in the reference.
    MatPtrs mats;
    for (int i = 0; i < 6; ++i) mats.m[i] = (const float*)d_in[4 + i];

    float* ws   = (float*)d_ws;
    float* z_re = ws;
    float* z_im = ws + (size_t)n;
    float* y_re = ws + (size_t)2 * n;
    float* y_im = ws + (size_t)3 * n;
    float* Gp   = ws + (size_t)4 * n;              // KSPLIT * 12 * n floats
    float* Kbuf = ws + (size_t)(4 + 12 * KSPLIT) * n;   // 12*n floats

    const int tiles = 6 * (n / 16);
    dim3 mvGrid((tiles + WAVES_PER_BLOCK - 1) / WAVES_PER_BLOCK, KSPLIT);
    const int L = n / KSPLIT;
    size_t smem = (size_t)(3 * L + 8) * sizeof(float);
    dim3 ewGrid((n + 255) / 256);

    hopf_init<<<ewGrid, 256, 0, stream>>>(zin_re, zin_im, z_re, z_im, y_re, y_im, n);

    const float h = (float)(6.283185307179586 / 96.0);
    static const float rows[6][6] = {
        {0.2f, 0.f, 0.f, 0.f, 0.f, 0.f},
        {3.f/40.f, 9.f/40.f, 0.f, 0.f, 0.f, 0.f},
        {44.f/45.f, -56.f/15.f, 32.f/9.f, 0.f, 0.f, 0.f},
        {19372.f/6561.f, -25360.f/2187.f, 64448.f/6561.f, -212.f/729.f, 0.f, 0.f},
        {9017.f/3168.f, -355.f/33.f, 46732.f/5247.f, 49.f/176.f, -5103.f/18656.f, 0.f},
        {35.f/384.f, 0.f, 500.f/1113.f, 125.f/192.f, -2187.f/6784.f, 11.f/84.f}
    };

    for (int step = 0; step < 96; ++step) {
        for (int st = 0; st < 6; ++st) {
            hopf_matvec<<<mvGrid, MV_BLOCK, smem, stream>>>(mats, y_re, y_im, Gp, n, tiles);
            const float* c = rows[st];
            hopf_dot_update<<<1, DU_BLOCK, 0, stream>>>(
                Gp, z_re, z_im, Kbuf, y_re, y_im, z_re, z_im,
                st, st + 1, c[0], c[1], c[2], c[3], c[4], c[5],
                h, (st == 5) ? 1 : 0, n);
        }
    }

    hopf_out<<<ewGrid, 256, 0, stream>>>(z_re, z_im, (float*)d_out, n);
}